// PostProcessing_41283225649325
// MI455X (gfx1250) — compile-verified
//
#include <hip/hip_runtime.h>
#include <stdint.h>

#define NUM_CLASSES 80
#define MAX_BOXES   20
#define SCORE_THR   0.6f
#define IOU_THR     0.5f
#define TOPK        256
#define CAP         8192
#define N_ANCHORS   100800

typedef __attribute__((ext_vector_type(2))) float v2f;
typedef __attribute__((ext_vector_type(8))) float v8f;

__device__ __forceinline__ float sigmoidf_fast(float x) {
    // v_rcp_f32-based sigmoid (1 TRANS + 1 TRANS instead of IEEE divide chain)
    return __builtin_amdgcn_rcpf(1.0f + __expf(-x));
}

// ---------------- zero per-class candidate counters ----------------
__global__ void k_init(int* __restrict__ candCnt) {
    int t = blockIdx.x * blockDim.x + threadIdx.x;
    if (t < NUM_CLASSES) candCnt[t] = 0;
}

// ---------------- fused decode + box-correct + threshold + append ----------------
// One thread per (cell, anchor). input_shape = (1280,1280), image = (720,1280):
//   min(input/image)=1 -> new_shape=(720,1280), offset=(0.21875,0), scale=(16/9,1).
// Class loop is reduced to load+compare by inverting the score threshold to a
// raw-logit threshold: conf*sigmoid(v) >= THR  <=>  v >= logit(THR/conf).
__global__ void k_decode(const float* __restrict__ feats, int gw, int gh,
                         float inv_gw, float inv_gh, int base,
                         float aw0, float ah0, float aw1, float ah1, float aw2, float ah2,
                         float* __restrict__ boxesOut, int* __restrict__ candCnt,
                         float* __restrict__ candScore, int* __restrict__ candIdx) {
    int tid = blockIdx.x * blockDim.x + threadIdx.x;
    int n = gw * gh * 3;
    if (tid >= n) return;
    int cell = tid / 3, a = tid - cell * 3;
    int cy = cell / gw, cx = cell - cy * gw;
    const float* p = feats + (size_t)cell * 255 + a * 85;
    float f0 = p[0], f1 = p[1], f2 = p[2], f3 = p[3], f4 = p[4];
    float aw = (a == 0) ? aw0 : ((a == 1) ? aw1 : aw2);
    float ah = (a == 0) ? ah0 : ((a == 1) ? ah1 : ah2);

    float x = (sigmoidf_fast(f0) + (float)cx) * inv_gw;
    float y = (sigmoidf_fast(f1) + (float)cy) * inv_gh;
    float w = __expf(f2) * aw * (1.0f / 1280.0f);
    float h = __expf(f3) * ah * (1.0f / 1280.0f);

    const float offY = 0.21875f, sclY = 1280.0f / 720.0f;
    float yc = (y - offY) * sclY;
    float hc = h * sclY;
    float y1 = (yc - 0.5f * hc) * 720.0f;
    float y2 = (yc + 0.5f * hc) * 720.0f;
    float x1 = (x - 0.5f * w) * 1280.0f;
    float x2 = (x + 0.5f * w) * 1280.0f;

    int gidx = base + tid;
    ((float4*)boxesOut)[gidx] = make_float4(y1, x1, y2, x2);

    float conf = sigmoidf_fast(f4);
    // conservative raw-logit prefilter threshold (exact re-check on passers)
    float z = SCORE_THR * __builtin_amdgcn_rcpf(conf);   // THR / conf
    float tlogit;
    if (z >= 1.0f) {
        tlogit = __builtin_inff();                        // conf < THR: nothing can pass
    } else {
        tlogit = __logf(z * __builtin_amdgcn_rcpf(1.0f - z)) - 1e-3f;
    }
    for (int c = 0; c < NUM_CLASSES; ++c) {
        float fv = p[5 + c];
        if (fv >= tlogit) {
            float s = conf * sigmoidf_fast(fv);
            if (s >= SCORE_THR) {
                int pos = atomicAdd(&candCnt[c], 1);
                if (pos < CAP) {
                    candScore[(size_t)c * CAP + pos] = s;
                    candIdx[(size_t)c * CAP + pos]   = gidx;
                }
            }
        }
    }
}

// ---------------- per-class top-256 via in-LDS bitonic sort (descending) ----------------
__global__ void k_topk(const int* __restrict__ candCnt,
                       const float* __restrict__ candScore, const int* __restrict__ candIdx,
                       float* __restrict__ selScore, int* __restrict__ selIdx) {
    __shared__ float ss[CAP];
    __shared__ int   si[CAP];
    int c = blockIdx.x;
    int tid = threadIdx.x;
    int n = candCnt[c]; if (n > CAP) n = CAP;
    for (int i = tid; i < CAP; i += 256) {
        if (i < n) { ss[i] = candScore[(size_t)c * CAP + i]; si[i] = candIdx[(size_t)c * CAP + i]; }
        else       { ss[i] = -1.0f;                          si[i] = 0; }
    }
    __syncthreads();
    for (int k = 2; k <= CAP; k <<= 1) {
        for (int j = k >> 1; j > 0; j >>= 1) {
            for (int i = tid; i < CAP; i += 256) {
                int ixj = i ^ j;
                if (ixj > i) {
                    bool desc = ((i & k) == 0);
                    float av = ss[i], bv = ss[ixj];
                    bool sw = desc ? (av < bv) : (av > bv);
                    if (sw) { ss[i] = bv; ss[ixj] = av; int tt = si[i]; si[i] = si[ixj]; si[ixj] = tt; }
                }
            }
            __syncthreads();
        }
    }
    if (tid < TOPK) {
        selScore[c * TOPK + tid] = ss[tid];
        selIdx[c * TOPK + tid]   = si[tid];
    }
}

// ---------------- per-class NMS: WMMA area-sum tiles + ballot suppression rows ----------------
__global__ void k_nms(const float* __restrict__ boxesG,
                      const float* __restrict__ selScoreG, const int* __restrict__ selIdxG,
                      float* __restrict__ outF) {
    __shared__ float selS[TOPK];
    __shared__ int   selI[TOPK];
    __shared__ float y1s[TOPK], x1s[TOPK], y2s[TOPK], x2s[TOPK], areaS[TOPK];
    __shared__ unsigned supM[TOPK * 8];   // 256 rows x 256-bit suppression mask
    __shared__ unsigned keepW[8];

    int c = blockIdx.x;
    int t = threadIdx.x;
    int lane = t & 31, wave = t >> 5;

    // Async global->LDS copy of this class's selected {score, idx} (ASYNCcnt path).
    {
        unsigned ldsS = (unsigned)(uintptr_t)&selS[t];
        const float* gS = selScoreG + c * TOPK + t;
        asm volatile("global_load_async_to_lds_b32 %0, %1, off" :: "v"(ldsS), "v"(gS) : "memory");
        unsigned ldsI = (unsigned)(uintptr_t)&selI[t];
        const int* gI = selIdxG + c * TOPK + t;
        asm volatile("global_load_async_to_lds_b32 %0, %1, off" :: "v"(ldsI), "v"(gI) : "memory");
        asm volatile("s_wait_asynccnt 0" ::: "memory");
    }
    __syncthreads();

    // Gather candidate boxes, compute areas into LDS (SoA).
    int idx = selI[t];
    float4 b = ((const float4*)boxesG)[idx];
    y1s[t] = b.x; x1s[t] = b.y; y2s[t] = b.z; x2s[t] = b.w;
    areaS[t] = (b.z - b.x) * (b.w - b.y);
    for (int i = t; i < TOPK * 8; i += 256) supM[i] = 0u;
    unsigned vb = __builtin_amdgcn_ballot_w32(selS[t] >= SCORE_THR);
    if (lane == 0) keepW[wave] = vb;
    __syncthreads();

    // 16x16 tiles of the 256x256 pair space; 8 waves x 32 tiles.
    // WMMA computes areaSum[i][j] = area_i + area_j via rank-2 product:
    //   A rows = [area_i, 1, 0, 0] (16x4), B cols = [1, area_j, 0, 0]^T (4x16).
    #pragma unroll 1
    for (int tile = wave; tile < 256; tile += 8) {
        int ti = tile >> 4, tj = tile & 15;
        int bi = ti * 16, bj = tj * 16;
        bool lo = lane < 16;
        int l16 = lane & 15;
        v2f A, B;
        A.x = lo ? areaS[bi + l16] : 0.0f;  // lanes 0-15: K=0 ; lanes 16-31: K=2 (zero)
        A.y = lo ? 1.0f            : 0.0f;  // lanes 0-15: K=1 ; lanes 16-31: K=3 (zero)
        B.x = lo ? 1.0f            : 0.0f;  // row K=0 ones ; row K=2 zero
        B.y = lo ? areaS[bj + l16] : 0.0f;  // row K=1 area_j ; row K=3 zero
        v8f zero = {};
        v8f areaSum = __builtin_amdgcn_wmma_f32_16x16x4_f32(
            false, A, false, B, (short)0, zero, false, false);

        int j = bj + l16;
        float jy1 = y1s[j], jx1 = x1s[j], jy2 = y2s[j], jx2 = x2s[j];
        unsigned word  = (unsigned)(tj >> 1);
        unsigned shift = (unsigned)(tj & 1) * 16u;
        #pragma unroll
        for (int r = 0; r < 8; ++r) {
            int i = bi + r + (lo ? 0 : 8);  // matches C/D VGPR layout: M = r + 8*(lane>=16)
            float ih = fminf(y2s[i], jy2) - fmaxf(y1s[i], jy1);
            float iw = fminf(x2s[i], jx2) - fmaxf(x1s[i], jx1);
            float inter = fmaxf(ih, 0.0f) * fmaxf(iw, 0.0f);
            // inter/(areaSum-inter+eps) > THR  <=>  inter > THR*(areaSum-inter+eps)
            // (denominator is always > 0: inter <= min(area_i, area_j))
            bool sup = (inter > IOU_THR * (areaSum[r] - inter + 1e-9f)) && (j > i);
            unsigned m = __builtin_amdgcn_ballot_w32(sup);
            if (lane == 0)  atomicOr(&supM[(bi + r) * 8 + word],     (m & 0xFFFFu) << shift);
            if (lane == 16) atomicOr(&supM[(bi + 8 + r) * 8 + word], (m >> 16)     << shift);
        }
    }
    __syncthreads();

    // Zero output slots, write class ids.
    if (t < MAX_BOXES) {
        int o = c * MAX_BOXES + t;
        outF[o * 4 + 0] = 0.0f; outF[o * 4 + 1] = 0.0f;
        outF[o * 4 + 2] = 0.0f; outF[o * 4 + 3] = 0.0f;
        outF[NUM_CLASSES * MAX_BOXES * 4 + o] = 0.0f;
        ((int*)outF)[NUM_CLASSES * MAX_BOXES * 5 + o] = c;
    }
    __syncthreads();

    // Serial greedy pass (2048 LDS word ops) + compaction of first 20 survivors.
    if (t == 0) {
        for (int i = 0; i < TOPK; ++i) {
            if ((keepW[i >> 5] >> (i & 31)) & 1u) {
                #pragma unroll
                for (int w2 = 0; w2 < 8; ++w2) keepW[w2] &= ~supM[i * 8 + w2];
            }
        }
        int nout = 0;
        for (int i = 0; i < TOPK && nout < MAX_BOXES; ++i) {
            if ((keepW[i >> 5] >> (i & 31)) & 1u) {
                int o = c * MAX_BOXES + nout;
                outF[o * 4 + 0] = y1s[i]; outF[o * 4 + 1] = x1s[i];
                outF[o * 4 + 2] = y2s[i]; outF[o * 4 + 3] = x2s[i];
                outF[NUM_CLASSES * MAX_BOXES * 4 + o] = selS[i];
                ++nout;
            }
        }
    }
}

extern "C" void kernel_launch(void* const* d_in, const int* in_sizes, int n_in,
                              void* d_out, int out_size, void* d_ws, size_t ws_size,
                              hipStream_t stream) {
    (void)in_sizes; (void)n_in; (void)out_size; (void)ws_size;
    const float* f0 = (const float*)d_in[0];   // 1x40x40x255
    const float* f1 = (const float*)d_in[1];   // 1x80x80x255
    const float* f2 = (const float*)d_in[2];   // 1x160x160x255
    float* out = (float*)d_out;

    char* ws = (char*)d_ws;
    float* boxes   = (float*)ws;                                          // 100800 * 4 f32
    int*   candCnt = (int*)(ws + (size_t)N_ANCHORS * 16);                 // 80 i32
    float* candSc  = (float*)(ws + (size_t)N_ANCHORS * 16 + 512);         // 80*CAP f32
    int*   candId  = (int*)((char*)candSc + (size_t)NUM_CLASSES * CAP * 4);
    float* selSc   = (float*)((char*)candId + (size_t)NUM_CLASSES * CAP * 4);
    int*   selId   = (int*)((char*)selSc + (size_t)NUM_CLASSES * TOPK * 4);

    k_init<<<1, 128, 0, stream>>>(candCnt);
    // scale 0: 40x40, anchors mask [6,7,8]
    k_decode<<<(4800 + 255) / 256, 256, 0, stream>>>(
        f0, 40, 40, 1.0f / 40.0f, 1.0f / 40.0f, 0,
        116.f, 90.f, 156.f, 198.f, 373.f, 326.f,
        boxes, candCnt, candSc, candId);
    // scale 1: 80x80, anchors mask [3,4,5]
    k_decode<<<(19200 + 255) / 256, 256, 0, stream>>>(
        f1, 80, 80, 1.0f / 80.0f, 1.0f / 80.0f, 4800,
        30.f, 61.f, 62.f, 45.f, 59.f, 119.f,
        boxes, candCnt, candSc, candId);
    // scale 2: 160x160, anchors mask [0,1,2]
    k_decode<<<(76800 + 255) / 256, 256, 0, stream>>>(
        f2, 160, 160, 1.0f / 160.0f, 1.0f / 160.0f, 24000,
        10.f, 13.f, 16.f, 30.f, 33.f, 23.f,
        boxes, candCnt, candSc, candId);
    k_topk<<<NUM_CLASSES, 256, 0, stream>>>(candCnt, candSc, candId, selSc, selId);
    k_nms<<<NUM_CLASSES, 256, 0, stream>>>(boxes, selSc, selId, out);
}